// _GCN_59313498358226
// MI455X (gfx1250) — compile-verified
//
#include <hip/hip_runtime.h>
#include <hip/hip_bf16.h>

typedef float v8f __attribute__((ext_vector_type(8)));
typedef float v2f __attribute__((ext_vector_type(2)));

#define D64 64

// ---------------- degree / norm precompute ----------------

__global__ __launch_bounds__(256) void k_zero(float* p, int n) {
    int i = blockIdx.x * 256 + threadIdx.x;
    if (i < n) p[i] = 0.0f;
}

__global__ __launch_bounds__(256) void k_deg(const int* __restrict__ col,
                                             const float* __restrict__ w,
                                             float* __restrict__ deg, int E) {
    int e = blockIdx.x * 256 + threadIdx.x;
    if (e < E) atomicAdd(&deg[col[e]], w[e]);
}

__global__ __launch_bounds__(256) void k_dinv(const float* __restrict__ deg,
                                              float* __restrict__ dinv, int n) {
    int i = blockIdx.x * 256 + threadIdx.x;
    if (i < n) {
        float d = deg[i] + 1.0f;          // self-loop fill_value = 1
        dinv[i] = rsqrtf(d);              // d > 0 always
    }
}

__global__ __launch_bounds__(256) void k_norm(const int* __restrict__ row,
                                              const int* __restrict__ col,
                                              const float* __restrict__ w,
                                              const float* __restrict__ dinv,
                                              float* __restrict__ norm, int E) {
    int e = blockIdx.x * 256 + threadIdx.x;
    if (e < E) norm[e] = dinv[row[e]] * w[e] * dinv[col[e]];
}

// ---------------- fp32 WMMA GEMM: T = H (Nx64) @ W (64x64) ----------------
// 256 threads = 8 waves; block covers 32 rows x 64 cols = 2x4 tiles of 16x16.
// Per wave: 16x V_WMMA_F32_16X16X4_F32 over K=64.

__global__ __launch_bounds__(256) void k_gemm64(const float* __restrict__ H,
                                                const float* __restrict__ W,
                                                float* __restrict__ T) {
    __shared__ float sW[D64 * D64];
    int tid = threadIdx.x;
    // stage W into LDS (4096 floats, 1024 float4 by 256 threads)
    for (int i = tid; i < 1024; i += 256)
        ((float4*)sW)[i] = ((const float4*)W)[i];
    __syncthreads();

    int wave = tid >> 5;            // 0..7
    int lane = tid & 31;
    int rowTile = wave >> 2;        // 0..1
    int colTile = wave & 3;         // 0..3
    int m0 = blockIdx.x * 32 + rowTile * 16;
    int n0 = colTile * 16;
    int mr = lane & 15;             // row within tile (A) / col within tile (B,D)
    int kb = (lane >> 4) * 2;       // lane-group K offset: 0 or 2
    int nc = n0 + mr;               // output column for B/D access

    const float* arow = H + (size_t)(m0 + mr) * D64;
    v8f acc = {};
#pragma unroll
    for (int k = 0; k < D64; k += 4) {
        v2f a = *(const v2f*)(arow + k + kb);           // A[m][k+kb], A[m][k+kb+1]
        v2f b;
        b.x = sW[(k + kb) * D64 + nc];                  // B[k+kb][n]
        b.y = sW[(k + kb + 1) * D64 + nc];              // B[k+kb+1][n]
        acc = __builtin_amdgcn_wmma_f32_16x16x4_f32(
            /*neg_a=*/false, a, /*neg_b=*/false, b,
            /*c_mod=*/(short)0, acc, /*reuse_a=*/false, /*reuse_b=*/false);
    }

    // D layout: VGPR r -> row (m0 + r + 8*(lane/16)), col nc
    int rbase = m0 + ((lane >> 4) << 3);
#pragma unroll
    for (int r = 0; r < 8; ++r)
        T[(size_t)(rbase + r) * D64 + nc] = acc[r];
}

// ---------------- per-layer message passing ----------------

// out[i][d] = b[d] + tmp[i][d] * dinv[i]^2   (bias + self-loop, no atomics)
__global__ __launch_bounds__(256) void k_init(const float* __restrict__ tmp,
                                              const float* __restrict__ bias,
                                              const float* __restrict__ dinv,
                                              float* __restrict__ out, int nd) {
    int i = blockIdx.x * 256 + threadIdx.x;
    if (i < nd) {
        float dv = dinv[i >> 6];
        out[i] = bias[i & 63] + tmp[i] * dv * dv;
    }
}

// out[col[e]][d] += tmp[row[e]][d] * norm[e]
__global__ __launch_bounds__(256) void k_scatter(const int* __restrict__ row,
                                                 const int* __restrict__ col,
                                                 const float* __restrict__ norm,
                                                 const float* __restrict__ tmp,
                                                 float* __restrict__ out, int E) {
    int gid = blockIdx.x * 256 + threadIdx.x;
    int e = gid >> 6;
    int d = gid & 63;
    if (e < E) {
        float v = tmp[(size_t)row[e] * D64 + d] * norm[e];
        atomicAdd(&out[(size_t)col[e] * D64 + d], v);
    }
}

__global__ __launch_bounds__(256) void k_relu(float* __restrict__ h, int nd) {
    int i = blockIdx.x * 256 + threadIdx.x;
    if (i < nd) h[i] = fmaxf(h[i], 0.0f);
}

// ---------------- driver ----------------

extern "C" void kernel_launch(void* const* d_in, const int* in_sizes, int n_in,
                              void* d_out, int out_size, void* d_ws, size_t ws_size,
                              hipStream_t stream) {
    const float* x    = (const float*)d_in[0];
    const int*   ei   = (const int*)d_in[1];     // [2, E]
    const float* ew   = (const float*)d_in[2];
    const float* Ws[3] = { (const float*)d_in[3], (const float*)d_in[5], (const float*)d_in[7] };
    const float* bs[3] = { (const float*)d_in[4], (const float*)d_in[6], (const float*)d_in[8] };

    const int N  = in_sizes[0] / D64;
    const int E  = in_sizes[2];
    const int ND = N * D64;
    const int* row = ei;
    const int* col = ei + E;

    float* deg  = (float*)d_ws;
    float* dinv = deg + N;
    float* norm = dinv + N;
    float* tmp  = norm + E;          // N*64 floats

    float* out = (float*)d_out;      // [x | h1 | h2 | h3], each N*64

    const int TB = 256;
    dim3 blk(TB);

    // ---- GCN normalization ----
    k_zero<<<dim3((N + TB - 1) / TB), blk, 0, stream>>>(deg, N);
    k_deg<<<dim3((E + TB - 1) / TB), blk, 0, stream>>>(col, ew, deg, E);
    k_dinv<<<dim3((N + TB - 1) / TB), blk, 0, stream>>>(deg, dinv, N);
    k_norm<<<dim3((E + TB - 1) / TB), blk, 0, stream>>>(row, col, ew, dinv, norm, E);

    // ---- copy x into output slot 0 ----
    hipMemcpyAsync(out, x, (size_t)ND * sizeof(float), hipMemcpyDeviceToDevice, stream);

    // ---- three GCN layers ----
    const float* h_in = x;
    for (int l = 0; l < 3; ++l) {
        float* h_out = out + (size_t)(l + 1) * ND;
        k_gemm64<<<dim3(N / 32), blk, 0, stream>>>(h_in, Ws[l], tmp);
        k_init<<<dim3((ND + TB - 1) / TB), blk, 0, stream>>>(tmp, bs[l], dinv, h_out, ND);
        k_scatter<<<dim3((E * 64 + TB - 1) / TB), blk, 0, stream>>>(row, col, norm, tmp, h_out, E);
        if (l < 2)
            k_relu<<<dim3((ND + TB - 1) / TB), blk, 0, stream>>>(h_out, ND);
        h_in = h_out;
    }
}